// FFTConvNet_81827716923396
// MI455X (gfx1250) — compile-verified
//
#include <hip/hip_runtime.h>

// ============================================================================
// FFTConvNet for MI455X (gfx1250, wave32, WMMA).
//
// Math: out[b,o] = bias[o] + Re(IFFT2( mask * sum_i FFT2(x)[(b+8)%16, i]
//                                             * FFT2(w)[(o+32)%64, i] ))
//   (all-dims fftshift: channel roll cancels in the contraction; batch/out-
//    channel rolls are involutions folded into indexing; mask -> only the
//    centered 64x64 frequency band survives.)
//
// Every transform is a GEMM vs the 128x128 DFT matrix F, executed with
// V_WMMA_F32_16X16X4_F32 (fp32 matrix pipe; ~19 GFLOP total vs ~0.4 GB of
// HBM traffic, i.e. near the 23.3 TB/s memory floor).
// ============================================================================

typedef __attribute__((ext_vector_type(2))) float v2f;
typedef __attribute__((ext_vector_type(8))) float v8f;

#define PI_F 3.14159265358979323846f

__device__ __forceinline__ v8f wmma4(v2f a, v2f b, v8f c) {
  // D = A(16x4) * B(4x16) + C, fp32.  f32 WMMA: no A/B neg (NEG=CNeg only).
  return __builtin_amdgcn_wmma_f32_16x16x4_f32(false, a, false, b, (short)0, c,
                                               false, false);
}

// band index (0..63, centered c = bi-32) -> source frequency index 0..127
__device__ __forceinline__ int bandmap(int bi) { return (bi + 96) & 127; }

// ---- fragment loaders (ISA 7.12.2 layouts, wave32) -------------------------
// A 16x4 tile: lanes 0-15 hold M, VGPR0/1 = K0/K1; lanes 16-31 = K2/K3.
__device__ __forceinline__ v2f fragA(const float* A, int lda, int m0, int k0) {
  int lane = threadIdx.x & 31;
  int m = m0 + (lane & 15);
  int kk = k0 + ((lane >> 4) << 1);
  v2f r; r.x = A[m * lda + kk]; r.y = A[m * lda + kk + 1]; return r;
}
// B 4x16 tile: row striped across lanes; VGPR q holds rows q and q+2.
__device__ __forceinline__ v2f fragB(const float* B, int ldb, int k0, int n0) {
  int lane = threadIdx.x & 31;
  int n = n0 + (lane & 15);
  int kk = k0 + ((lane >> 4) << 1);
  v2f r; r.x = B[kk * ldb + n]; r.y = B[(kk + 1) * ldb + n]; return r;
}
// B[k][n] = F[k*128 + bandmap(n)]      (forward, band columns)
__device__ __forceinline__ v2f fragB_bc(const float* F, int k0, int n0) {
  int lane = threadIdx.x & 31;
  int col = bandmap(n0 + (lane & 15));
  int kk = k0 + ((lane >> 4) << 1);
  v2f r; r.x = F[kk * 128 + col]; r.y = F[(kk + 1) * 128 + col]; return r;
}
// A[m][k] = F[bandmap(m)*128 + k]      (forward, band rows)
__device__ __forceinline__ v2f fragA_br(const float* F, int m0, int k0) {
  int lane = threadIdx.x & 31;
  int row = bandmap(m0 + (lane & 15));
  int kk = k0 + ((lane >> 4) << 1);
  v2f r; r.x = F[row * 128 + kk]; r.y = F[row * 128 + kk + 1]; return r;
}
// A[m][k] = F[m*128 + bandmap(k)]      (inverse stage 1)
__device__ __forceinline__ v2f fragA_kmap(const float* F, int m0, int k0) {
  int lane = threadIdx.x & 31;
  int m = m0 + (lane & 15);
  int kk = k0 + ((lane >> 4) << 1);
  v2f r; r.x = F[m * 128 + bandmap(kk)]; r.y = F[m * 128 + bandmap(kk + 1)];
  return r;
}
// B[k][n] = F[bandmap(k)*128 + n]      (inverse stage 2)
__device__ __forceinline__ v2f fragB_kmap(const float* F, int k0, int n0) {
  int lane = threadIdx.x & 31;
  int n = n0 + (lane & 15);
  int kk = k0 + ((lane >> 4) << 1);
  v2f r; r.x = F[bandmap(kk) * 128 + n]; r.y = F[bandmap(kk + 1) * 128 + n];
  return r;
}

// ---- K0: DFT twiddle tables F[j,k] = exp(-2*pi*i*j*k/128) ------------------
__global__ void k0_tables(float* __restrict__ Ftab) {
  for (int t = threadIdx.x + blockIdx.x * blockDim.x; t < 16384;
       t += blockDim.x * gridDim.x) {
    int j = t >> 7, k = t & 127;
    float ang = -2.f * PI_F * (float)((j * k) & 127) / 128.f;
    float s, c;
    sincosf(ang, &s, &c);
    Ftab[t] = c;            // Re
    Ftab[16384 + t] = s;    // Im
  }
}

// ---- K1: forward FFT2 of x, band-restricted, masked, batch-roll folded -----
// One block per (b,i) image; 8 waves. T1 = X * F[:,band] then Xb = F[band,:]*T1.
__global__ void k1_fwd(const float* __restrict__ x, const float* __restrict__ Ftab,
                       float* __restrict__ XbR, float* __restrict__ XbI) {
  __shared__ float T1r[128 * 64];
  __shared__ float T1i[128 * 64];
  const float* Fre = Ftab;
  const float* Fim = Ftab + 16384;
  int blk = blockIdx.x;
  int b = blk >> 6, ci = blk & 63;
  const float* Ximg = x + ((size_t)blk << 14);
  int wave = threadIdx.x >> 5;
  int lane = threadIdx.x & 31;

  // Stage 1: T1[h][vv] = sum_w X[h][w] * F[w][bandcol(vv)]   (real * complex)
  for (int t = wave; t < 32; t += 8) {                   // 8x4 tiles of 16x16
    int m0 = (t >> 2) << 4, n0 = (t & 3) << 4;
    v8f aR = {}, aI = {};
    for (int k = 0; k < 128; k += 4) {
      v2f a  = fragA(Ximg, 128, m0, k);
      v2f br = fragB_bc(Fre, k, n0);
      v2f bi = fragB_bc(Fim, k, n0);
      aR = wmma4(a, br, aR);
      aI = wmma4(a, bi, aI);
    }
    int n = n0 + (lane & 15);
    int mb = m0 + ((lane >> 4) << 3);
#pragma unroll
    for (int r = 0; r < 8; ++r) {
      T1r[(mb + r) * 64 + n] = aR[r];
      T1i[(mb + r) * 64 + n] = aI[r];
    }
  }
  __syncthreads();

  // Stage 2: Xb[uu][vv] = mask * sum_h F[bandrow(uu)][h] * T1[h][vv]
  int bdst = (b + 8) & 15;                               // fftshift batch roll
  for (int t = wave; t < 16; t += 8) {                   // 4x4 tiles
    int m0 = (t >> 2) << 4, n0 = (t & 3) << 4;
    v8f cR = {}, cI = {};
    for (int k = 0; k < 128; k += 4) {
      v2f ar = fragA_br(Fre, m0, k);
      v2f ai = fragA_br(Fim, m0, k);
      v2f br = fragB(T1r, 64, k, n0);
      v2f bi = fragB(T1i, 64, k, n0);
      cR = wmma4(ar, br, cR);
      v2f ani; ani.x = -ai.x; ani.y = -ai.y;             // f32 WMMA lacks A-neg
      cR = wmma4(ani, bi, cR);
      cI = wmma4(ar, bi, cI);
      cI = wmma4(ai, br, cI);
    }
    int n = n0 + (lane & 15);
    int mb = m0 + ((lane >> 4) << 3);
    int dv = n - 32;
#pragma unroll
    for (int r = 0; r < 8; ++r) {
      int uu = mb + r;
      int du = uu - 32;
      float msk = (du * du + dv * dv <= 900) ? 1.f : 0.f;  // radius 30 low-pass
      size_t off = ((size_t)(uu * 64 + n) << 10) + (size_t)(bdst << 6) + ci;
      XbR[off] = cR[r] * msk;
      XbI[off] = cI[r] * msk;
    }
  }
}

// ---- K2: weight DFT at band frequencies (direct 9-tap), o-roll folded ------
__global__ void k2_wdft(const float* __restrict__ w, float* __restrict__ WbR,
                        float* __restrict__ WbI) {
  int f = blockIdx.x;                 // 4096 band frequencies
  int u = bandmap(f >> 6), v = bandmap(f & 63);
  float cy[3], sy[3], cx[3], sx[3];
#pragma unroll
  for (int t = 0; t < 3; ++t) {
    sincosf(-2.f * PI_F * (float)((u * t) & 127) / 128.f, &sy[t], &cy[t]);
    sincosf(-2.f * PI_F * (float)((v * t) & 127) / 128.f, &sx[t], &cx[t]);
  }
  float* outR = WbR + ((size_t)f << 12);
  float* outI = WbI + ((size_t)f << 12);
  for (int p = threadIdx.x; p < 4096; p += blockDim.x) {  // p = i*64 + o'
    int i = p >> 6, op = p & 63;
    int osrc = (op + 32) & 63;                            // fftshift o roll
    const float* wk = w + (size_t)(osrc * 64 + i) * 9;
    float ar = 0.f, ai = 0.f;
#pragma unroll
    for (int ky = 0; ky < 3; ++ky)
#pragma unroll
      for (int kx = 0; kx < 3; ++kx) {
        float tr = cy[ky] * cx[kx] - sy[ky] * sx[kx];
        float ti = cy[ky] * sx[kx] + sy[ky] * cx[kx];
        float wv = wk[ky * 3 + kx];
        ar += wv * tr;
        ai += wv * ti;
      }
    outR[p] = ar;
    outI[p] = ai;
  }
}

// ---- K3: per-frequency complex channel-mix GEMM  Z(16x64)=X(16x64)*W(64x64)
__global__ void k3_mix(const float* __restrict__ XbR, const float* __restrict__ XbI,
                       const float* __restrict__ WbR, const float* __restrict__ WbI,
                       float* __restrict__ YbR, float* __restrict__ YbI) {
  int f = blockIdx.x;
  const float* AR = XbR + ((size_t)f << 10);
  const float* AI = XbI + ((size_t)f << 10);
  const float* BR = WbR + ((size_t)f << 12);
  const float* BI = WbI + ((size_t)f << 12);
  __builtin_prefetch(BR, 0, 1);       // global_prefetch_b8 on the hot W panel
  __builtin_prefetch(BI, 0, 1);
  int wave = threadIdx.x >> 5;        // 4 waves, one 16x16 N-tile each
  int lane = threadIdx.x & 31;
  int n0 = wave << 4;
  v8f cR = {}, cI = {};
  for (int k = 0; k < 64; k += 4) {
    v2f ar = fragA(AR, 64, 0, k);
    v2f ai = fragA(AI, 64, 0, k);
    v2f br = fragB(BR, 64, k, n0);
    v2f bi = fragB(BI, 64, k, n0);
    cR = wmma4(ar, br, cR);
    v2f ani; ani.x = -ai.x; ani.y = -ai.y;
    cR = wmma4(ani, bi, cR);
    cI = wmma4(ar, bi, cI);
    cI = wmma4(ai, br, cI);
  }
  int n = n0 + (lane & 15);
  int mb = (lane >> 4) << 3;
#pragma unroll
  for (int r = 0; r < 8; ++r) {
    size_t off = ((size_t)((mb + r) * 64 + n) << 12) + f;  // Yb[b][o][f]
    YbR[off] = cR[r];
    YbI[off] = cI[r];
  }
}

// ---- K4: pruned inverse FFT2 (+ bias) per (b,o) image ----------------------
// S = conj(F)[:,band] * Y ; out = Re(S * conj(F)[band,:]) / 16384 + bias
__global__ void k4_inv(const float* __restrict__ YbR, const float* __restrict__ YbI,
                       const float* __restrict__ Ftab, const float* __restrict__ bias,
                       float* __restrict__ out) {
  __shared__ float Sr[128 * 64];
  __shared__ float Si[128 * 64];
  const float* Fre = Ftab;
  const float* Fim = Ftab + 16384;
  int blk = blockIdx.x;
  const float* Yr = YbR + ((size_t)blk << 12);
  const float* Yi = YbI + ((size_t)blk << 12);
  int wave = threadIdx.x >> 5;
  int lane = threadIdx.x & 31;

  // Stage 1 (complex): S[m][vv] = sum_uu conjF[m][bandrow(uu)] * Y[uu][vv]
  for (int t = wave; t < 32; t += 8) {
    int m0 = (t >> 2) << 4, n0 = (t & 3) << 4;
    v8f cR = {}, cI = {};
    for (int k = 0; k < 64; k += 4) {
      v2f ar = fragA_kmap(Fre, m0, k);
      v2f ai = fragA_kmap(Fim, m0, k);   // conj -> (ar - i*ai)
      v2f br = fragB(Yr, 64, k, n0);
      v2f bi = fragB(Yi, 64, k, n0);
      cR = wmma4(ar, br, cR);            // Re += ar*br + ai*bi
      cR = wmma4(ai, bi, cR);
      cI = wmma4(ar, bi, cI);            // Im += ar*bi - ai*br
      v2f ani; ani.x = -ai.x; ani.y = -ai.y;
      cI = wmma4(ani, br, cI);
    }
    int n = n0 + (lane & 15);
    int mb = m0 + ((lane >> 4) << 3);
#pragma unroll
    for (int r = 0; r < 8; ++r) {
      Sr[(mb + r) * 64 + n] = cR[r];
      Si[(mb + r) * 64 + n] = cI[r];
    }
  }
  __syncthreads();

  // Stage 2 (real part only): out[m][n] = sum_vv Sr*Fre + Si*Fim  (conj twice)
  float bo = bias[blk & 63];
  float* oimg = out + ((size_t)blk << 14);
  for (int t = wave; t < 64; t += 8) {                   // 8x8 tiles
    int m0 = (t >> 3) << 4, n0 = (t & 7) << 4;
    v8f c = {};
    for (int k = 0; k < 64; k += 4) {
      v2f sr = fragA(Sr, 64, m0, k);
      v2f si = fragA(Si, 64, m0, k);
      v2f br = fragB_kmap(Fre, k, n0);
      v2f bi = fragB_kmap(Fim, k, n0);
      c = wmma4(sr, br, c);
      c = wmma4(si, bi, c);
    }
    int n = n0 + (lane & 15);
    int mb = m0 + ((lane >> 4) << 3);
#pragma unroll
    for (int r = 0; r < 8; ++r)
      oimg[(mb + r) * 128 + n] = c[r] * (1.f / 16384.f) + bo;
  }
}

// ---- launch ----------------------------------------------------------------
extern "C" void kernel_launch(void* const* d_in, const int* in_sizes, int n_in,
                              void* d_out, int out_size, void* d_ws, size_t ws_size,
                              hipStream_t stream) {
  (void)in_sizes; (void)n_in; (void)out_size; (void)ws_size;
  const float* x    = (const float*)d_in[0];   // [16,64,128,128]
  const float* w    = (const float*)d_in[1];   // [64,64,3,3]
  const float* bias = (const float*)d_in[2];   // [64]
  float* out = (float*)d_out;                  // [16,64,128,128]

  float* ws   = (float*)d_ws;                  // ~201.5 MB of fp32 scratch
  float* Ftab = ws;                            // 2 * 16384
  float* XbR  = ws + 32768;                    // [4096 f][16 b][64 i]
  float* XbI  = XbR + 4194304;
  float* WbR  = XbI + 4194304;                 // [4096 f][64 i][64 o]
  float* WbI  = WbR + 16777216;
  float* YbR  = WbI + 16777216;                // [16 b][64 o][4096 f]
  float* YbI  = YbR + 4194304;

  k0_tables<<<16, 256, 0, stream>>>(Ftab);
  k1_fwd  <<<1024, 256, 0, stream>>>(x, Ftab, XbR, XbI);
  k2_wdft <<<4096, 256, 0, stream>>>(w, WbR, WbI);
  k3_mix  <<<4096, 128, 0, stream>>>(XbR, XbI, WbR, WbI, YbR, YbI);
  k4_inv  <<<1024, 256, 0, stream>>>(YbR, YbI, Ftab, bias, out);
}